// AudioWaveAugment_26259430048670
// MI455X (gfx1250) — compile-verified
//
#include <hip/hip_runtime.h>
#include <cstdint>

#define B_N 128
#define T_N 320000
#define HALO 16
#define TILE 1024
#define NTHR 256
#define RED_BLOCKS 16
#define NELEM (TILE + 2 * HALO)   // 1056 floats = 264 x 16B chunks

typedef float v4f __attribute__((ext_vector_type(4)));

// ---------- Pass 1a: per-(batch, slice) partial sums (deterministic, no atomics) ----------
__global__ __launch_bounds__(NTHR)
void red_partial(const float* __restrict__ x, double* __restrict__ part) {
    const int b = blockIdx.y;
    const v4f* x4 = (const v4f*)(x + (size_t)b * T_N);
    const int nf4 = T_N / 4;  // 80000
    float s = 0.f, ss = 0.f;
    for (int i = blockIdx.x * NTHR + threadIdx.x; i < nf4; i += RED_BLOCKS * NTHR) {
        v4f v = x4[i];   // regular-temporal: keep x resident in the 192MB L2 for pass 2
        s  += v.x + v.y + v.z + v.w;
        ss += v.x * v.x + v.y * v.y + v.z * v.z + v.w * v.w;
    }
    __shared__ double sh0[NTHR], sh1[NTHR];
    sh0[threadIdx.x] = (double)s;
    sh1[threadIdx.x] = (double)ss;
    __syncthreads();
    for (int off = NTHR / 2; off > 0; off >>= 1) {
        if ((int)threadIdx.x < off) {
            sh0[threadIdx.x] += sh0[threadIdx.x + off];
            sh1[threadIdx.x] += sh1[threadIdx.x + off];
        }
        __syncthreads();
    }
    if (threadIdx.x == 0) {
        part[(b * RED_BLOCKS + blockIdx.x) * 2 + 0] = sh0[0];
        part[(b * RED_BLOCKS + blockIdx.x) * 2 + 1] = sh1[0];
    }
}

// ---------- Pass 1b: finalize std(x) per batch (ddof=1) ----------
__global__ void red_final(const double* __restrict__ part, double* __restrict__ stdx) {
    const int b = threadIdx.x;
    if (b >= B_N) return;
    double s = 0.0, ss = 0.0;
    for (int i = 0; i < RED_BLOCKS; ++i) {
        s  += part[(b * RED_BLOCKS + i) * 2 + 0];
        ss += part[(b * RED_BLOCKS + i) * 2 + 1];
    }
    const double N = (double)T_N;
    double var = (ss - s * s / N) / (N - 1.0);
    stdx[b] = var > 0.0 ? sqrt(var) : 0.0;
}

// ---------- Pass 2: gain + noise + box filter (low/high pass), streaming ----------
__global__ __launch_bounds__(NTHR)
void augment(const float* __restrict__ x, const float* __restrict__ noise,
             const float* __restrict__ gains, const float* __restrict__ nscales,
             const float* __restrict__ do_gain, const float* __restrict__ do_noise,
             const float* __restrict__ do_filter, const float* __restrict__ low_coin,
             const int* __restrict__ halves, const double* __restrict__ stdx,
             float* __restrict__ out) {
    const int b = blockIdx.y;
    const int tileStart = blockIdx.x * TILE;

    // Per-batch uniform parameters (wave-uniform scalar path)
    const float g = (do_gain[b] < 0.7f) ? gains[b] : 1.0f;   // GAIN_PROB
    float std1 = g * (float)stdx[b];                         // std(x*g) = g*std(x)
    if (std1 < 1e-4f) std1 = 1e-4f;
    const float s = (do_noise[b] < 0.5f) ? std1 * nscales[b] : 0.0f;  // NOISE_PROB
    const bool apply_filter = do_filter[b] < 0.35f;          // FILTER_PROB
    const bool use_low = low_coin[b] < 0.5f;
    const int h = halves[b];                                 // 2..16
    const float kf = (float)(2 * h + 1);

    __shared__ float xs[NELEM];
    __shared__ float ns[NELEM];

    const size_t base = (size_t)b * T_N;
    const int lo = tileStart - HALO;                         // global index of lds[0]
    const bool edge = (lo < 0) || (tileStart + TILE + HALO > T_N);

    if (!edge) {
        // CDNA5 async global->LDS DMA: 16B per lane, tracked with ASYNCcnt.
        // x: regular-temporal (reuse from pass 1 via L2).  noise: non-temporal (streamed once).
        const uint64_t gx = (uint64_t)(uintptr_t)(x + base + lo);
        const uint64_t gn = (uint64_t)(uintptr_t)(noise + base + lo);
        const uint32_t lx = (uint32_t)(uintptr_t)&xs[0];
        const uint32_t ln = (uint32_t)(uintptr_t)&ns[0];
        const int c = threadIdx.x;
        asm volatile("global_load_async_to_lds_b128 %0, %1, off"
                     :: "v"(lx + (uint32_t)c * 16u), "v"(gx + (uint64_t)c * 16u) : "memory");
        asm volatile("global_load_async_to_lds_b128 %0, %1, off th:TH_LOAD_NT"
                     :: "v"(ln + (uint32_t)c * 16u), "v"(gn + (uint64_t)c * 16u) : "memory");
        if (c < (NELEM / 4 - NTHR)) {   // 8 leftover 16B chunks
            const int c2 = c + NTHR;
            asm volatile("global_load_async_to_lds_b128 %0, %1, off"
                         :: "v"(lx + (uint32_t)c2 * 16u), "v"(gx + (uint64_t)c2 * 16u) : "memory");
            asm volatile("global_load_async_to_lds_b128 %0, %1, off th:TH_LOAD_NT"
                         :: "v"(ln + (uint32_t)c2 * 16u), "v"(gn + (uint64_t)c2 * 16u) : "memory");
        }
        asm volatile("s_wait_asynccnt 0" ::: "memory");
    } else {
        // Edge tiles: explicit zero padding matches count_include_pad avg_pool semantics.
        for (int i = threadIdx.x; i < NELEM; i += NTHR) {
            const int gt = lo + i;
            const bool ok = (gt >= 0) && (gt < T_N);
            xs[i] = ok ? x[base + gt] : 0.0f;
            ns[i] = ok ? __builtin_nontemporal_load(noise + base + gt) : 0.0f;
        }
    }
    __syncthreads();

    const int o = threadIdx.x * 4;          // 4 consecutive outputs per thread
    const int t0 = tileStart + o;
    if (t0 >= T_N) return;
    const int cb = o + HALO;

    // x2 = g*x + s*noise  (center values)
    float x2v[4];
#pragma unroll
    for (int j = 0; j < 4; ++j) x2v[j] = g * xs[cb + j] + s * ns[cb + j];

    v4f res;
    if (apply_filter) {
        // lowpass(g*x + s*n) = (g*Sum(x) + s*Sum(n)) / k  -> sliding window sums
        float Sx = 0.f, Sn = 0.f;
        for (int d = cb - h; d <= cb + h; ++d) { Sx += xs[d]; Sn += ns[d]; }
        float low = (g * Sx + s * Sn) / kf;
        res.x = use_low ? low : x2v[0] - low;
#pragma unroll
        for (int j = 1; j < 4; ++j) {
            Sx += xs[cb + h + j] - xs[cb - h + j - 1];
            Sn += ns[cb + h + j] - ns[cb - h + j - 1];
            low = (g * Sx + s * Sn) / kf;
            float r = use_low ? low : x2v[j] - low;
            if (j == 1) res.y = r; else if (j == 2) res.z = r; else res.w = r;
        }
    } else {
        res.x = x2v[0]; res.y = x2v[1]; res.z = x2v[2]; res.w = x2v[3];
    }
    // NT store: don't let the output stream evict x from L2
    __builtin_nontemporal_store(res, (v4f*)(out + base + t0));
}

extern "C" void kernel_launch(void* const* d_in, const int* in_sizes, int n_in,
                              void* d_out, int out_size, void* d_ws, size_t ws_size,
                              hipStream_t stream) {
    const float* x         = (const float*)d_in[0];
    const float* gains     = (const float*)d_in[1];
    const float* nscales   = (const float*)d_in[2];
    const float* noise     = (const float*)d_in[3];
    const float* do_gain   = (const float*)d_in[4];
    const float* do_noise  = (const float*)d_in[5];
    const float* do_filter = (const float*)d_in[6];
    const float* low_coin  = (const float*)d_in[7];
    const int*   halves    = (const int*)d_in[8];
    float* out = (float*)d_out;

    double* part = (double*)d_ws;                                   // B*16*2 doubles
    double* stdx = part + (size_t)B_N * RED_BLOCKS * 2;             // B doubles (~33 KB total)

    red_partial<<<dim3(RED_BLOCKS, B_N), NTHR, 0, stream>>>(x, part);
    red_final<<<1, B_N, 0, stream>>>(part, stdx);

    const int tiles = (T_N + TILE - 1) / TILE;                      // 313
    augment<<<dim3(tiles, B_N), NTHR, 0, stream>>>(x, noise, gains, nscales, do_gain,
                                                   do_noise, do_filter, low_coin,
                                                   halves, stdx, out);
}